// GroupItN_867583394680
// MI455X (gfx1250) — compile-verified
//
#include <hip/hip_runtime.h>
#include <hip/hip_bf16.h>

// ---------------------------------------------------------------------------
// Group whitening (IterNorm-style) for MI455X / gfx1250.
//   x: (64, 256, 56, 56) fp32, G=32 groups, m = 8*3136 = 25088 per group.
// Bandwidth-bound (~616 MB traffic -> ~26us at 23.3 TB/s):
//   K1: zero scratch accumulators
//   K2: streaming Gram (32x32/sample) + row sums; async global->LDS tile
//       fills (GLOBAL_LOAD_ASYNC_TO_LDS_B128) + WMMA f32 16x16x4
//   K3: per-sample Newton-Schulz (5 iters of 32x32 matmuls), WMMA f32 16x16x4
//   K4: streaming apply y = wm@x - (wm@mu), *weight + bias (async tile fills)
// ---------------------------------------------------------------------------

typedef __attribute__((ext_vector_type(2))) float v2f;
typedef __attribute__((ext_vector_type(8))) float v8f;

#define NSAMP 64
#define NGRP 32
#define MCOLS 25088          // (256/32)*56*56
#define HW 3136              // 56*56
#define TILE_COLS 128
#define NTILES 196           // 25088 / 128
#define LDS_STRIDE 132       // row stride: 528B = 33*16 -> 16B-aligned rows,
                             // 132 % 64 = 4 -> conflict-free fragment reads

// ---- CDNA5 async global->LDS copy (ASYNCcnt path, bypasses VGPRs) ----------
// One wave moves one 512B row: 32 lanes x 16B.
__device__ __forceinline__ void giw_async_row_b128(unsigned lds_byte,
                                                   const float* gsrc) {
    asm volatile("global_load_async_to_lds_b128 %0, %1, off"
                 :: "v"(lds_byte), "v"(gsrc)
                 : "memory");
}
__device__ __forceinline__ void giw_wait_async0() {
#if __has_builtin(__builtin_amdgcn_s_wait_asynccnt)
    __builtin_amdgcn_s_wait_asynccnt(0);
#else
    asm volatile("s_wait_asynccnt 0" ::: "memory");
#endif
}
// low 32 bits of a generic pointer to __shared__ == LDS byte offset
__device__ __forceinline__ unsigned giw_lds_addr(const void* p) {
    return (unsigned)(unsigned long long)p;
}

// ---------------- K1: zero the atomic-accumulated scratch -------------------
__global__ void giw_zero_kernel(float* __restrict__ p, int n) {
    int i = blockIdx.x * blockDim.x + threadIdx.x;
    if (i < n) p[i] = 0.0f;
}

// --------- WMMA fragment helpers (V_WMMA_F32_16X16X4_F32 layouts) ----------
// A 16x4 (MxK): lanes 0-15: M=lane, VGPR0/1 = K0/K1; lanes 16-31: M=lane-16,
// VGPR0/1 = K2/K3.  B 4x16 (KxN): VGPR v, lanes 0-15 = row K=v, lanes 16-31 =
// row K=v+2.  C/D 16x16: lane<16: VGPR v=(M=v,N=lane); lane>=16: (M=v+8,N=lane-16).

__device__ __forceinline__ v8f mm_tile_acc(const float* __restrict__ A,
                                           const float* __restrict__ B,
                                           int ti, int tj, int lane, v8f acc) {
    #pragma unroll
    for (int k0 = 0; k0 < 8; ++k0) {
        const int kc = k0 * 4;
        v2f a, b;
        if (lane < 16) {
            a.x = A[(ti * 16 + lane) * 33 + kc + 0];
            a.y = A[(ti * 16 + lane) * 33 + kc + 1];
            b.x = B[(kc + 0) * 33 + tj * 16 + lane];
            b.y = B[(kc + 1) * 33 + tj * 16 + lane];
        } else {
            const int l = lane - 16;
            a.x = A[(ti * 16 + l) * 33 + kc + 2];
            a.y = A[(ti * 16 + l) * 33 + kc + 3];
            b.x = B[(kc + 2) * 33 + tj * 16 + l];
            b.y = B[(kc + 3) * 33 + tj * 16 + l];
        }
        acc = __builtin_amdgcn_wmma_f32_16x16x4_f32(false, a, false, b,
                                                    (short)0, acc, false, false);
    }
    return acc;
}

__device__ __forceinline__ void mm_tile_store(float* __restrict__ D, v8f acc,
                                              int ti, int tj, int lane) {
    const int colL = lane & 15;
    const int radd = (lane < 16) ? 0 : 8;
    #pragma unroll
    for (int v = 0; v < 8; ++v)
        D[(ti * 16 + v + radd) * 33 + tj * 16 + colL] = acc[v];
}

// ---------------- K2: Gram matrix (uncentered) + row sums -------------------
// grid = (14, 64); 8 waves/block: waves own (16x16 tile, column half) and
// carry WMMA accumulators across all column tiles -> few final atomics.
__global__ __launch_bounds__(256) void giw_gram_kernel(
        const float* __restrict__ x, float* __restrict__ gram,
        float* __restrict__ sums) {
    __shared__ __align__(16) float tile[NGRP * LDS_STRIDE];
    const int n = blockIdx.y;
    const int t = threadIdx.x;
    const int lane = t & 31;
    const int wave = t >> 5;
    const int tsub = wave & 3;       // which 16x16 tile of the 32x32 Gram
    const int ti = tsub >> 1, tj = tsub & 1;
    const int khalf = wave >> 2;     // cols 0-63 or 64-127 of each LDS tile
    const float* xs = x + (size_t)n * (NGRP * (size_t)MCOLS);

    v8f acc = {};
    float rowsum = 0.0f;

    for (int tIdx = blockIdx.x; tIdx < NTILES; tIdx += gridDim.x) {
        const int col0 = tIdx * TILE_COLS;
        __syncthreads();   // previous tile fully consumed before overwrite
        // async fill: each wave DMAs 4 rows (512B each) straight into LDS
        #pragma unroll
        for (int i = 0; i < 4; ++i) {
            const int r = wave * 4 + i;
            const unsigned lds = giw_lds_addr(&tile[r * LDS_STRIDE + lane * 4]);
            giw_async_row_b128(lds, xs + (size_t)r * MCOLS + col0 + lane * 4);
        }
        giw_wait_async0();
        __syncthreads();
        if (t < 32) {      // wave 0 also gathers row sums (for the mean)
            float s = 0.0f;
            for (int c = 0; c < TILE_COLS; ++c) s += tile[t * LDS_STRIDE + c];
            rowsum += s;
        }
        // Gram tile: A = rows ti*16.., B = rows tj*16.. of the same tile
        const int kbase = khalf * 16;
        for (int k0 = kbase; k0 < kbase + 16; ++k0) {
            const int kc = k0 * 4;
            v2f a, b;
            if (lane < 16) {
                a.x = tile[(ti * 16 + lane) * LDS_STRIDE + kc + 0];
                a.y = tile[(ti * 16 + lane) * LDS_STRIDE + kc + 1];
                b.x = tile[(tj * 16 + lane) * LDS_STRIDE + kc + 0];
                b.y = tile[(tj * 16 + lane) * LDS_STRIDE + kc + 1];
            } else {
                const int l = lane - 16;
                a.x = tile[(ti * 16 + l) * LDS_STRIDE + kc + 2];
                a.y = tile[(ti * 16 + l) * LDS_STRIDE + kc + 3];
                b.x = tile[(tj * 16 + l) * LDS_STRIDE + kc + 2];
                b.y = tile[(tj * 16 + l) * LDS_STRIDE + kc + 3];
            }
            acc = __builtin_amdgcn_wmma_f32_16x16x4_f32(false, a, false, b,
                                                        (short)0, acc, false, false);
        }
    }
    // reduce: 28 atomics per Gram element total across the grid
    float* gn = gram + (size_t)n * 1024;
    const int colL = lane & 15;
    const int radd = (lane < 16) ? 0 : 8;
    #pragma unroll
    for (int v = 0; v < 8; ++v) {
        const int row = ti * 16 + v + radd;
        const int col = tj * 16 + colL;
        atomicAdd(&gn[row * 32 + col], acc[v]);
    }
    if (t < 32) atomicAdd(&sums[n * 32 + t], rowsum);
}

// ---------------- K3: per-sample Newton-Schulz (tiny) -----------------------
// grid = 64 blocks of 128 threads (4 waves = 2x2 tiles of 32x32).
__global__ __launch_bounds__(128) void giw_ns_kernel(
        const float* __restrict__ gram, const float* __restrict__ sums,
        float* __restrict__ wm_out, float* __restrict__ v_out, float m_inv) {
    __shared__ float sig[32 * 33], P[32 * 33], T1[32 * 33], T2[32 * 33];
    __shared__ float mu[32];
    __shared__ float trInv_s;
    const int n = blockIdx.x;
    const int t = threadIdx.x;
    const int lane = t & 31;
    const int wave = t >> 5;
    const int ti = wave >> 1, tj = wave & 1;
    const float* gn = gram + (size_t)n * 1024;

    if (t < 32) mu[t] = sums[n * 32 + t] * m_inv;
    __syncthreads();
    // sigma = Gram/m - mu*mu^T
    for (int i = t; i < 1024; i += 128) {
        const int r = i >> 5, c = i & 31;
        sig[r * 33 + c] = gn[i] * m_inv - mu[r] * mu[c];
    }
    __syncthreads();
    if (t == 0) {
        float tr = 0.0f;
        for (int g = 0; g < 32; ++g) tr += sig[g * 33 + g];
        trInv_s = 1.0f / tr;
    }
    __syncthreads();
    const float trInv = trInv_s;
    for (int i = t; i < 1024; i += 128) {
        const int r = i >> 5, c = i & 31;
        sig[r * 33 + c] *= trInv;                 // sigma_n
        P[r * 33 + c] = (r == c) ? 1.0f : 0.0f;   // P = I
    }
    __syncthreads();

    for (int it = 0; it < 5; ++it) {
        v8f a1 = mm_tile_acc(P, P, ti, tj, lane, v8f{});     // T1 = P@P
        mm_tile_store(T1, a1, ti, tj, lane);
        __syncthreads();
        v8f a2 = mm_tile_acc(T1, P, ti, tj, lane, v8f{});    // T2 = T1@P
        mm_tile_store(T2, a2, ti, tj, lane);
        __syncthreads();
        v8f a3 = mm_tile_acc(T2, sig, ti, tj, lane, v8f{});  // P3@sigma_n
        {   // P = 1.5P - 0.5*a3  (each wave owns its tile -> in-place safe)
            const int colL = lane & 15;
            const int radd = (lane < 16) ? 0 : 8;
            #pragma unroll
            for (int v = 0; v < 8; ++v) {
                const int r = ti * 16 + v + radd, c = tj * 16 + colL;
                P[r * 33 + c] = 1.5f * P[r * 33 + c] - 0.5f * a3[v];
            }
        }
        __syncthreads();
    }

    // wm = P * sqrt(trInv);  v = wm @ mu (fold the centering into apply pass)
    const float s = sqrtf(trInv);
    for (int i = t; i < 1024; i += 128) {
        const int r = i >> 5, c = i & 31;
        const float w = P[r * 33 + c] * s;
        wm_out[(size_t)n * 1024 + i] = w;
        T1[r * 33 + c] = w;
    }
    __syncthreads();
    if (t < 32) {
        float acc = 0.0f;
        for (int h = 0; h < 32; ++h) acc += T1[t * 33 + h] * mu[h];
        v_out[n * 32 + t] = acc;
    }
}

// ---------------- K4: streaming apply: y = wm@x - v, *weight + bias ---------
// grid = (196, 64), 256 threads; read + write 205 MB each -> bandwidth-bound.
__global__ __launch_bounds__(256) void giw_apply_kernel(
        const float* __restrict__ x, const float* __restrict__ wm,
        const float* __restrict__ vvec, const float* __restrict__ weight,
        const float* __restrict__ bias, float* __restrict__ y) {
    __shared__ __align__(16) float tile[NGRP * LDS_STRIDE];
    __shared__ float wms[1024];
    __shared__ float vs[32];
    __shared__ float wch[256], bch[256];
    const int n = blockIdx.y;
    const int t = threadIdx.x;
    const int lane = t & 31;
    const int wave = t >> 5;
    const float* xs = x + (size_t)n * (NGRP * (size_t)MCOLS);
    float* ys = y + (size_t)n * (NGRP * (size_t)MCOLS);

    // async DMA of the x tile (each wave: 4 rows of 512B)
    const int col0 = blockIdx.x * TILE_COLS;
    #pragma unroll
    for (int i = 0; i < 4; ++i) {
        const int r = wave * 4 + i;
        const unsigned lds = giw_lds_addr(&tile[r * LDS_STRIDE + lane * 4]);
        giw_async_row_b128(lds, xs + (size_t)r * MCOLS + col0 + lane * 4);
    }
    // small per-sample/per-channel params via normal loads
    for (int i = t; i < 1024; i += 256) wms[i] = wm[(size_t)n * 1024 + i];
    if (t < 32) vs[t] = vvec[n * 32 + t];
    wch[t] = weight[t];
    bch[t] = bias[t];
    giw_wait_async0();
    __syncthreads();

    const int col = t & 127;
    const int gbase = (t >> 7) * 16;          // each thread: 16 groups, 1 col
    const int colg = col0 + col;
    const int cg = colg / HW;                 // channel-within-group 0..7
    float xv[32];
    #pragma unroll
    for (int h = 0; h < 32; ++h) xv[h] = tile[h * LDS_STRIDE + col];
    #pragma unroll
    for (int gi = 0; gi < 16; ++gi) {
        const int g = gbase + gi;
        float acc = -vs[g];
        #pragma unroll
        for (int h = 0; h < 32; ++h) acc += wms[g * 32 + h] * xv[h];
        const int ch = g * 8 + cg;
        ys[(size_t)g * MCOLS + colg] = acc * wch[ch] + bch[ch];
    }
}

// ---------------------------------------------------------------------------
extern "C" void kernel_launch(void* const* d_in, const int* in_sizes, int n_in,
                              void* d_out, int out_size, void* d_ws, size_t ws_size,
                              hipStream_t stream) {
    (void)in_sizes; (void)n_in; (void)out_size; (void)ws_size;
    const float* x      = (const float*)d_in[0];
    const float* weight = (const float*)d_in[1];
    const float* bias   = (const float*)d_in[2];
    float* out = (float*)d_out;
    float* ws  = (float*)d_ws;

    // scratch layout (floats): gram[64*1024] | sums[64*32] | wm[64*1024] | v[64*32]
    float* gram = ws;
    float* sums = ws + 65536;
    float* wm   = ws + 67584;
    float* vv   = ws + 133120;

    const int nzero = 67584;  // gram + sums are atomically accumulated
    giw_zero_kernel<<<(nzero + 255) / 256, 256, 0, stream>>>(ws, nzero);
    giw_gram_kernel<<<dim3(14, NSAMP), 256, 0, stream>>>(x, gram, sums);
    giw_ns_kernel<<<NSAMP, 128, 0, stream>>>(gram, sums, wm, vv, 1.0f / (float)MCOLS);
    giw_apply_kernel<<<dim3(NTILES, NSAMP), 256, 0, stream>>>(x, wm, vv, weight, bias, out);
}